// AbstractLeakyReluLess_46222438039763
// MI455X (gfx1250) — compile-verified
//
#include <hip/hip_runtime.h>

// CDNA5 / gfx1250: wave32, WMMA f32 16x16x4 (A=16x4, B=4x16, C/D=16x16 over 8 VGPRs).
typedef __attribute__((ext_vector_type(2))) float v2f;
typedef __attribute__((ext_vector_type(8))) float v8f;

#define NNEUR  4096
#define MINP   4096
#define NSLOPE 0.01f
#define BLOCK  256
#define WAVES  8              // 256 threads / wave32
#define KSEG   (MINP / WAVES) // 512 K per wave

// Reduced math:
//   y1[i] = sum_j W[i,j]*mid[j],  y2[i] = sum_j |W[i,j]|*rad[j]
//   lower[i] = lw*(y1-y2+b),  upper[i] = uw*(y1+y2+b) + ub_bias
__global__ __launch_bounds__(BLOCK)
void abs_lrelu_backsub_kernel(const float* __restrict__ lbnd,
                              const float* __restrict__ ubnd,
                              const float* __restrict__ alpha,
                              const float* __restrict__ W,
                              const float* __restrict__ bias,
                              const float* __restrict__ l0,
                              const float* __restrict__ u0,
                              float* __restrict__ out)
{
    __shared__ float s_mid[MINP];
    __shared__ float s_rad[MINP];
    __shared__ float s_red[WAVES][2][16];

    const int tid = threadIdx.x;

    // Stage mid/rad into LDS once per block (32 KB; WGP has 320 KB).
    for (int j = tid; j < MINP; j += BLOCK) {
        float lo = l0[j], hi = u0[j];
        s_mid[j] = 0.5f * (hi + lo);
        s_rad[j] = 0.5f * (hi - lo);
    }
    __syncthreads();

    const int wave  = tid >> 5;          // 0..7 (wave32)
    const int lane  = tid & 31;
    const int khalf = (lane >> 4) << 1;  // 0 for lanes 0-15, 2 for lanes 16-31
    const int row_base = blockIdx.x << 4;
    const int row   = row_base + (lane & 15);
    const float* __restrict__ wrow = W + (size_t)row * MINP;

    // Four independent accumulator chains keep the matrix pipe busy.
    v8f acc1a = {}; v8f acc1b = {}; v8f acc2a = {}; v8f acc2b = {};

    const int kbeg = wave * KSEG + khalf;

    // Constant trip count -> wave-uniform scalar loop (no EXEC-masked divergent
    // loop idiom; EXEC provably all-ones around every WMMA).
    #pragma unroll 8
    for (int it = 0; it < KSEG / 8; ++it) {
        const int k = kbeg + (it << 3);
        // B operand: lane n (n<16) holds W[row_base+n, k+{0,1}], lane n+16 holds k+{2,3}.
        v2f wA = *(const v2f*)(wrow + k);
        v2f wB = *(const v2f*)(wrow + k + 4);
        // A operand: every logical A-row = the same 4-vector chunk (lanes 0-15 get
        // mid[k],mid[k+1]; lanes 16-31 get mid[k+2],mid[k+3]) -> D rows all equal y.
        v2f mA = *(const v2f*)(s_mid + k);
        v2f mB = *(const v2f*)(s_mid + k + 4);
        v2f rA = *(const v2f*)(s_rad + k);
        v2f rB = *(const v2f*)(s_rad + k + 4);
        v2f aA; aA.x = __builtin_fabsf(wA.x); aA.y = __builtin_fabsf(wA.y);
        v2f aB; aB.x = __builtin_fabsf(wB.x); aB.y = __builtin_fabsf(wB.y);

        // D = A x B + C  (args: neg_a, A, neg_b, B, c_mod, C, reuse_a, reuse_b)
        acc1a = __builtin_amdgcn_wmma_f32_16x16x4_f32(false, mA, false, wA, (short)0, acc1a, false, false);
        acc2a = __builtin_amdgcn_wmma_f32_16x16x4_f32(false, rA, false, aA, (short)0, acc2a, false, false);
        acc1b = __builtin_amdgcn_wmma_f32_16x16x4_f32(false, mB, false, wB, (short)0, acc1b, false, false);
        acc2b = __builtin_amdgcn_wmma_f32_16x16x4_f32(false, rB, false, aB, (short)0, acc2b, false, false);
    }

    // D VGPR0, lanes 0-15: y for rows row_base+0..15 (all D rows identical by construction).
    float y1 = acc1a[0] + acc1b[0];
    float y2 = acc2a[0] + acc2b[0];
    if (lane < 16) {
        s_red[wave][0][lane] = y1;
        s_red[wave][1][lane] = y2;
    }
    __syncthreads();

    if (tid < 16) {
        float t1 = 0.f, t2 = 0.f;
        #pragma unroll
        for (int w = 0; w < WAVES; ++w) { t1 += s_red[w][0][tid]; t2 += s_red[w][1][tid]; }

        const int i = row_base + tid;
        const float lbv = lbnd[i], ubv = ubnd[i], av = alpha[i], bv = bias[i];
        float lw = NSLOPE, uw = NSLOPE, ubb = 0.f;
        if (lbv > 0.f) {
            lw = 1.f; uw = 1.f;
        } else if (lbv < 0.f && ubv > 0.f) {
            float slope = (ubv - NSLOPE * lbv) / (ubv - lbv);
            uw  = slope;
            ubb = (NSLOPE - slope) * lbv;
            lw  = fminf(fmaxf(av, NSLOPE), 1.f);
        }
        out[i]         = lw * (t1 - t2 + bv);          // lower
        out[NNEUR + i] = uw * (t1 + t2 + bv) + ubb;    // upper
    }
}

extern "C" void kernel_launch(void* const* d_in, const int* in_sizes, int n_in,
                              void* d_out, int out_size, void* d_ws, size_t ws_size,
                              hipStream_t stream) {
    (void)in_sizes; (void)n_in; (void)d_ws; (void)ws_size; (void)out_size;
    const float* lbnd  = (const float*)d_in[0];
    const float* ubnd  = (const float*)d_in[1];
    const float* alpha = (const float*)d_in[2];
    const float* W     = (const float*)d_in[3];
    const float* bias  = (const float*)d_in[4];
    const float* l0    = (const float*)d_in[5];
    const float* u0    = (const float*)d_in[6];
    float* out = (float*)d_out;

    dim3 grid(NNEUR / 16);   // 256 strips of 16 rows
    dim3 block(BLOCK);       // 8 wave32s
    abs_lrelu_backsub_kernel<<<grid, block, 0, stream>>>(lbnd, ubnd, alpha, W, bias, l0, u0, out);
}